// GNNwPOSENC_74259984548270
// MI455X (gfx1250) — compile-verified
//
#include <hip/hip_runtime.h>
#include <hip/hip_bf16.h>

typedef __attribute__((ext_vector_type(2))) float v2f;
typedef __attribute__((ext_vector_type(8))) float v8f;

#define DIVC 1.0e-4f

// ---------------------------------------------------------------------------
// Node feature construction: h0[i][0..23] = [glyph emb(16), xe(2), ye(2), agent(4)]
// Reproduces the reference's concatenate(axis=0).reshape(-1,2) interleave.
// ---------------------------------------------------------------------------
__device__ __forceinline__ float posenc_val(const int* __restrict__ x, int j, int col, int N) {
    if (j < N) return sinf((float)x[(size_t)j * 4 + col] * DIVC);
    return cosf((float)x[(size_t)(j - N) * 4 + col] * DIVC);
}

__global__ void embed_kernel(const int* __restrict__ x,
                             const float* __restrict__ eg,   // (6000,16)
                             const float* __restrict__ ea,   // (2,4)
                             float* __restrict__ h,          // (N,24)
                             int N) {
    int i = blockIdx.x * blockDim.x + threadIdx.x;
    if (i >= N) return;
    const int* xi = x + (size_t)i * 4;
    float* hp = h + (size_t)i * 24;
    int gl = xi[0], ag = xi[3];
    const float* ge = eg + (size_t)gl * 16;
    #pragma unroll
    for (int c = 0; c < 16; c++) hp[c] = ge[c];
    int j0 = 2 * i, j1 = 2 * i + 1;
    hp[16] = posenc_val(x, j0, 1, N);
    hp[17] = posenc_val(x, j1, 1, N);
    hp[18] = posenc_val(x, j0, 2, N);
    hp[19] = posenc_val(x, j1, 2, N);
    const float* ae = ea + (size_t)ag * 4;
    hp[20] = ae[0]; hp[21] = ae[1]; hp[22] = ae[2]; hp[23] = ae[3];
}

__global__ void edge_init_kernel(const int* __restrict__ ei, int* __restrict__ src,
                                 int* __restrict__ dst, float* __restrict__ em, int E) {
    int e = blockIdx.x * blockDim.x + threadIdx.x;
    if (e >= E) return;
    src[e] = ei[e];
    dst[e] = ei[(size_t)E + e];
    em[e] = 1.0f;
}

__global__ void fill_float_kernel(float* __restrict__ p, float v, int n) {
    int i = blockIdx.x * blockDim.x + threadIdx.x;
    if (i < n) p[i] = v;
}
__global__ void fill_int_kernel(int* __restrict__ p, int v, int n) {
    int i = blockIdx.x * blockDim.x + threadIdx.x;
    if (i < n) p[i] = v;
}

// deg[d] += emask[e]  (then finalized to rsqrt(deg+1))
__global__ void deg_edge_kernel(const int* __restrict__ dst, const float* __restrict__ em,
                                float* __restrict__ deg, int E) {
    int e = blockIdx.x * blockDim.x + threadIdx.x;
    if (e >= E) return;
    float m = em[e];
    if (m > 0.0f) atomicAdd(&deg[dst[e]], m);
}
__global__ void deg_fin_kernel(float* __restrict__ deg, int N) {
    int i = blockIdx.x * blockDim.x + threadIdx.x;
    if (i < N) deg[i] = rsqrtf(deg[i] + 1.0f);
}

// ---------------------------------------------------------------------------
// Dense h @ W via V_WMMA_F32_16X16X4_F32.  One wave owns a 16-row tile of h.
// A frag (16x4 f32): lanes 0-15 -> K=4s+{0,1}, lanes 16-31 -> K=4s+{2,3}
// B frag (4x16 f32): lanes 0-15 -> rows K=4s+{0,1} at N=lane,
//                    lanes 16-31 -> rows K=4s+{2,3} at N=lane-16
// D (16x16 f32): VGPR j -> M=j (lanes 0-15), M=j+8 (lanes 16-31), N=lane&15
// N is always a multiple of 16 in this pipeline, so EXEC stays all-ones.
// ---------------------------------------------------------------------------
__global__ void matmul_wmma_kernel(const float* __restrict__ h, const float* __restrict__ W,
                                   float* __restrict__ out, int N, int K, int relu) {
    int wave = (blockIdx.x * blockDim.x + threadIdx.x) >> 5;
    int lane = threadIdx.x & 31;
    int tiles = N >> 4;
    if (wave >= tiles) return;
    int m = lane & 15;
    int hi = lane >> 4;       // 0 or 1
    int koff = hi << 1;       // 0 or 2
    int row = (wave << 4) + m;
    v8f c = {};
    int steps = K >> 2;
    for (int s = 0; s < steps; s++) {
        int kb = (s << 2) + koff;
        const float* hp = h + (size_t)row * K + kb;
        float a0 = hp[0], a1 = hp[1];
        if (relu) { a0 = fmaxf(a0, 0.0f); a1 = fmaxf(a1, 0.0f); }
        v2f a; a.x = a0; a.y = a1;
        v2f b; b.x = W[kb * 16 + m]; b.y = W[(kb + 1) * 16 + m];
        c = __builtin_amdgcn_wmma_f32_16x16x4_f32(false, a, false, b, (short)0, c, false, false);
    }
    float* op = out + (((size_t)wave << 4) + ((size_t)hi << 3)) * 16 + m;
    #pragma unroll
    for (int j = 0; j < 8; j++) op[(size_t)j * 16] = c[j];
}

// agg[i] = hW[i] * dinv[i]^2 + bias   (self-loop term, initializes accumulator)
__global__ void self_init_kernel(const float* __restrict__ hW, const float* __restrict__ dinv,
                                 const float* __restrict__ bias, float* __restrict__ agg, int N) {
    int i = blockIdx.x * blockDim.x + threadIdx.x;
    if (i >= N * 16) return;
    int node = i >> 4, c = i & 15;
    float di = dinv[node];
    agg[i] = hW[i] * di * di + bias[c];
}

// agg[dst] += hW[src] * dinv[src]*dinv[dst]*emask   (atomic scatter over edges)
__global__ void edge_agg_kernel(const int* __restrict__ src, const int* __restrict__ dst,
                                const float* __restrict__ em, const float* __restrict__ dinv,
                                const float* __restrict__ hW, float* __restrict__ agg, int E) {
    int e = blockIdx.x * blockDim.x + threadIdx.x;
    if (e >= E) return;
    float m = em[e];
    if (m <= 0.5f) return;
    int s = src[e], d = dst[e];
    float nrm = dinv[s] * dinv[d] * m;
    const float4* hv = (const float4*)(hW + (size_t)s * 16);
    float* ap = agg + (size_t)d * 16;
    #pragma unroll
    for (int q = 0; q < 4; q++) {
        float4 v = hv[q];
        atomicAdd(ap + 4 * q + 0, v.x * nrm);
        atomicAdd(ap + 4 * q + 1, v.y * nrm);
        atomicAdd(ap + 4 * q + 2, v.z * nrm);
        atomicAdd(ap + 4 * q + 3, v.w * nrm);
    }
}

// ---------------------------------------------------------------------------
// Top-k pooling: one workgroup per graph; in-LDS bitonic sort of (score, idx),
// descending score, ascending index tie-break (matches jax.lax.top_k).
// M = npg padded to a power of two (<= 4096).
// ---------------------------------------------------------------------------
__global__ void topk_pool_kernel(const float* __restrict__ h, const float* __restrict__ w,
                                 float* __restrict__ hnew, int* __restrict__ newpos,
                                 int npg, int M, int k) {
    __shared__ float ssc[4096];
    __shared__ int   sidx[4096];
    __shared__ float s_inv;
    int g = blockIdx.x, tid = threadIdx.x, T = blockDim.x;
    if (tid == 0) {
        float s = 0.0f;
        #pragma unroll
        for (int c = 0; c < 16; c++) { float v = w[c]; s += v * v; }
        s_inv = 1.0f / sqrtf(s);
    }
    __syncthreads();
    float inv = s_inv;
    size_t base = (size_t)g * npg;
    for (int i = tid; i < M; i += T) {
        if (i < npg) {
            const float* hp = h + (base + i) * 16;
            float d = 0.0f;
            #pragma unroll
            for (int c = 0; c < 16; c++) d += hp[c] * w[c];
            ssc[i] = d * inv;
            sidx[i] = i;
        } else {
            ssc[i] = -3.0e38f;
            sidx[i] = 0x7fffffff;
        }
    }
    for (int size = 2; size <= M; size <<= 1) {
        for (int stride = size >> 1; stride > 0; stride >>= 1) {
            __syncthreads();
            for (int i = tid; i < M; i += T) {
                int j = i ^ stride;
                if (j > i) {
                    bool desc = ((i & size) == 0);
                    float si = ssc[i], sj = ssc[j];
                    int ii = sidx[i], ij = sidx[j];
                    bool iFirst = (si > sj) || (si == sj && ii < ij);
                    bool doSwap = desc ? (!iFirst) : iFirst;
                    if (doSwap) {
                        ssc[i] = sj; ssc[j] = si;
                        sidx[i] = ij; sidx[j] = ii;
                    }
                }
            }
        }
    }
    __syncthreads();
    for (int j = tid; j < k; j += T) {
        int l = sidx[j];
        float t = tanhf(ssc[j]);
        const float* op = h + (base + l) * 16;
        float* np = hnew + ((size_t)g * k + j) * 16;
        #pragma unroll
        for (int c = 0; c < 16; c++) np[c] = op[c] * t;
        newpos[base + l] = g * k + j;
    }
}

__global__ void edge_remap_kernel(int* __restrict__ src, int* __restrict__ dst,
                                  float* __restrict__ em, const int* __restrict__ newpos, int E) {
    int e = blockIdx.x * blockDim.x + threadIdx.x;
    if (e >= E) return;
    int s = src[e], d = dst[e];
    float m = em[e];
    int ns = newpos[s], nd = newpos[d];
    bool valid = (ns >= 0) && (nd >= 0) && (m > 0.5f);
    src[e] = valid ? ns : 0;
    dst[e] = valid ? nd : 0;
    em[e] = valid ? 1.0f : 0.0f;
}

__global__ void meanpool_kernel(const float* __restrict__ h, float* __restrict__ pooled, int k) {
    __shared__ float s[256];
    int g = blockIdx.x, tid = threadIdx.x;
    int c = tid & 15, q = tid >> 4;
    float sum = 0.0f;
    for (int j = q; j < k; j += 16) sum += h[((size_t)g * k + j) * 16 + c];
    s[tid] = sum;
    __syncthreads();
    if (tid < 16) {
        float t = 0.0f;
        #pragma unroll
        for (int q2 = 0; q2 < 16; q2++) t += s[q2 * 16 + tid];
        pooled[g * 16 + tid] = t / (float)k;
    }
}

__global__ void final_linear_kernel(const float* __restrict__ pooled, const float* __restrict__ W,
                                    const float* __restrict__ b, float* __restrict__ out) {
    int i = blockIdx.x * blockDim.x + threadIdx.x;
    if (i >= 32 * 23) return;
    int g = i / 23, o = i % 23;
    float acc = b[o];
    #pragma unroll
    for (int c = 0; c < 16; c++) acc += pooled[g * 16 + c] * W[c * 23 + o];
    out[i] = acc;
}

// ---------------------------------------------------------------------------
extern "C" void kernel_launch(void* const* d_in, const int* in_sizes, int n_in,
                              void* d_out, int out_size, void* d_ws, size_t ws_size,
                              hipStream_t stream) {
    const int N0 = 32 * 4096;      // 131072
    const int E  = N0 * 16;        // 2097152
    const int NPG = 4096;
    const int K1 = 2458, K2 = 1475, K3 = 885;   // ceil(0.6*{4096,2458,1475})

    const int*   x       = (const int*)d_in[0];
    const int*   ei      = (const int*)d_in[1];
    const float* eg      = (const float*)d_in[3];
    const float* ea      = (const float*)d_in[4];
    const float* firstW  = (const float*)d_in[5];
    const float* firstb  = (const float*)d_in[6];
    const float* Ws      = (const float*)d_in[7];
    const float* bs      = (const float*)d_in[8];
    const float* poolws  = (const float*)d_in[9];
    const float* linW    = (const float*)d_in[10];
    const float* linb    = (const float*)d_in[11];
    float* out = (float*)d_out;

    // Carve workspace (256B aligned slabs)
    uint8_t* wsb = (uint8_t*)d_ws;
    size_t off = 0;
    auto carve = [&](size_t bytes) -> void* {
        void* p = wsb + off;
        off += (bytes + 255) & ~(size_t)255;
        return p;
    };
    float* bufA   = (float*)carve(sizeof(float) * (size_t)N0 * 24);
    float* bufB   = (float*)carve(sizeof(float) * (size_t)N0 * 24);
    int*   ssrc   = (int*)carve(sizeof(int) * (size_t)E);
    int*   sdst   = (int*)carve(sizeof(int) * (size_t)E);
    float* sem    = (float*)carve(sizeof(float) * (size_t)E);
    float* dinv   = (float*)carve(sizeof(float) * (size_t)N0);
    int*   npos   = (int*)carve(sizeof(int) * (size_t)N0);
    float* pooled = (float*)carve(sizeof(float) * 32 * 16);
    (void)ws_size; (void)in_sizes; (void)n_in; (void)out_size;

    auto cdiv = [](int a, int b) { return (a + b - 1) / b; };

    auto compute_deg = [&](int N) {
        fill_float_kernel<<<cdiv(N, 256), 256, 0, stream>>>(dinv, 0.0f, N);
        deg_edge_kernel<<<cdiv(E, 256), 256, 0, stream>>>(sdst, sem, dinv, E);
        deg_fin_kernel<<<cdiv(N, 256), 256, 0, stream>>>(dinv, N);
    };
    auto run_gcn = [&](float* h, float* tmp, const float* W, const float* b,
                       int N, int K, int relu) {
        int tiles = N / 16;
        matmul_wmma_kernel<<<cdiv(tiles, 8), 256, 0, stream>>>(h, W, tmp, N, K, relu);
        self_init_kernel<<<cdiv(N * 16, 256), 256, 0, stream>>>(tmp, dinv, b, h, N);
        edge_agg_kernel<<<cdiv(E, 256), 256, 0, stream>>>(ssrc, sdst, sem, dinv, tmp, h, E);
    };

    // --- build features + mutable edge copies ---
    embed_kernel<<<cdiv(N0, 256), 256, 0, stream>>>(x, eg, ea, bufA, N0);
    edge_init_kernel<<<cdiv(E, 256), 256, 0, stream>>>(ei, ssrc, sdst, sem, E);

    float* h = bufA;
    float* t = bufB;

    // --- first GCN (K=24), then block 0 (5 layers) ---
    compute_deg(N0);
    run_gcn(h, t, firstW, firstb, N0, 24, 0);
    for (int li = 0; li < 5; li++)
        run_gcn(h, t, Ws + (size_t)li * 256, bs + (size_t)li * 16, N0, 16, 1);

    // --- pool 0 ---
    fill_int_kernel<<<cdiv(N0, 256), 256, 0, stream>>>(npos, -1, N0);
    topk_pool_kernel<<<32, 1024, 0, stream>>>(h, poolws + 0, t, npos, NPG, 4096, K1);
    edge_remap_kernel<<<cdiv(E, 256), 256, 0, stream>>>(ssrc, sdst, sem, npos, E);
    { float* tmp = h; h = t; t = tmp; }
    int N1 = 32 * K1;

    // --- block 1 ---
    compute_deg(N1);
    for (int li = 5; li < 10; li++)
        run_gcn(h, t, Ws + (size_t)li * 256, bs + (size_t)li * 16, N1, 16, 1);

    // --- pool 1 ---
    fill_int_kernel<<<cdiv(N1, 256), 256, 0, stream>>>(npos, -1, N1);
    topk_pool_kernel<<<32, 1024, 0, stream>>>(h, poolws + 16, t, npos, K1, 4096, K2);
    edge_remap_kernel<<<cdiv(E, 256), 256, 0, stream>>>(ssrc, sdst, sem, npos, E);
    { float* tmp = h; h = t; t = tmp; }
    int N2 = 32 * K2;

    // --- block 2 ---
    compute_deg(N2);
    for (int li = 10; li < 15; li++)
        run_gcn(h, t, Ws + (size_t)li * 256, bs + (size_t)li * 16, N2, 16, 1);

    // --- pool 2 (remapped edges unused afterwards -> skip edge_remap) ---
    fill_int_kernel<<<cdiv(N2, 256), 256, 0, stream>>>(npos, -1, N2);
    topk_pool_kernel<<<32, 1024, 0, stream>>>(h, poolws + 32, t, npos, K2, 2048, K3);
    { float* tmp = h; h = t; t = tmp; }

    // --- readout ---
    meanpool_kernel<<<32, 256, 0, stream>>>(h, pooled, K3);
    final_linear_kernel<<<cdiv(32 * 23, 256), 256, 0, stream>>>(pooled, linW, linb, out);
}